// SetConv2dDecoder_75943611728138
// MI455X (gfx1250) — compile-verified
//
#include <hip/hip_runtime.h>

// Problem constants (match reference: B_=4, C=8, N=1024, M=1024, DX=1)
#define BATCH 4
#define CCH   8
#define NN    1024
#define MM    1024

typedef __attribute__((ext_vector_type(2))) float v2f;
typedef __attribute__((ext_vector_type(8))) float v8f;

// column swizzle per K-pair-row: multiple of 4, period 8 rows -> breaks store
// bank conflicts, keeps float4 contiguity, uniform across a wave per kk-step
__device__ __forceinline__ constexpr int SW(int kp) { return ((kp >> 1) & 3) << 2; }

// ---------------------------------------------------------------------------
// 1) RBF weights: w[b,n,m] = exp(-0.5 * (x[b,n]-xz[m])^2 / exp(2*log_scale))
// ---------------------------------------------------------------------------
__global__ __launch_bounds__(256) void rbf_weights_kernel(
    const float* __restrict__ x, const float* __restrict__ xz,
    const float* __restrict__ log_scale, float* __restrict__ w) {
  const int idx = blockIdx.x * blockDim.x + threadIdx.x;  // over B*N*M
  const float inv_var = expf(-2.0f * log_scale[0]);
  const int m  = idx & (MM - 1);
  const int bn = idx >> 10;           // M == 1024
  const float d = x[bn] - xz[m];
  w[idx] = expf(-0.5f * d * d * inv_var);
}

// ---------------------------------------------------------------------------
// 2) Batched fp32 WMMA GEMM:  C = A(MxK) * B(KxN)   (TRANSB=1: B(k,n)=Bg[n,k])
//    - 128x128 block tile, BK=32, 256 threads = 8 wave32 (2x4), 64x32 per wave
//    - LDS K-pair-interleaved: elem (k,c) at [k>>1]*P2 + 2*(c^SW(k>>1)) + (k&1)
//      -> each WMMA fragment is ONE aligned ds_load_b64 (conflict-free)
//    - double-buffered LDS + register-staged global loads (1 barrier/iter)
// ---------------------------------------------------------------------------
template <int TRANSB>
__global__ __launch_bounds__(256) void wmma_gemm_f32(
    const float* __restrict__ Agbase, const float* __restrict__ Bgbase,
    float* __restrict__ Cgbase, int lda, int ldb, int ldc, int K,
    long aBStride, int aBDiv, long bBStride, long cBStride) {
  constexpr int P2 = 288;                   // floats per pair-row (256 + 32 pad), ≡32 mod 64
  __shared__ float As[2][16 * P2];          // [kpair][2*col interleaved]
  __shared__ float Bs[2][16 * P2];

  const int bz = blockIdx.z;
  const float* Ag = Agbase + (long)(bz / aBDiv) * aBStride;
  const float* Bg = Bgbase + (long)bz * bBStride;
  float*       Cg = Cgbase + (long)bz * cBStride;

  const int tileM = blockIdx.y * 128;
  const int tileN = blockIdx.x * 128;

  const int t    = threadIdx.x;
  const int lane = t & 31;
  const int wid  = t >> 5;
  const int wm   = wid >> 2;                // 0..1  (64-row half)
  const int wn   = wid & 3;                 // 0..3  (32-col quarter)
  const int l15  = lane & 15;
  const int lhi  = lane >> 4;               // 0/1: hi half-wave uses K+2,K+3

  // ---- loader mappings ----
  const int k4  = (t & 7) * 4;              // K offset of this thread's float4
  const int kp0 = (t & 7) * 2;              // first K-pair-row it writes
  const int rl  = t >> 3;                   // col (m or n) base, 0..31 (+32*p)
  const int n4  = (t & 31) * 4;             // float4 col offset (TRANSB=0 path)
  const int kr  = t >> 5;                   // k-pair row (TRANSB=0 path), 0..7 (+8*p)

  v8f acc[4][2];
#pragma unroll
  for (int i = 0; i < 4; ++i)
#pragma unroll
    for (int j = 0; j < 2; ++j) { v8f zz = {}; acc[i][j] = zz; }

  // ---- stage next tile's global data in registers ----
  float4 ra[4], rb0[2], rb1[2];

  auto loadA = [&](int k0) {
#pragma unroll
    for (int p = 0; p < 4; ++p)
      ra[p] = *(const float4*)(Ag + (long)(tileM + rl + 32 * p) * lda + k0 + k4);
  };
  auto storeA = [&](int buf) {
    const int s = SW(kp0);                  // SW(kp0) == SW(kp0+1), kp0 even
#pragma unroll
    for (int p = 0; p < 4; ++p) {
      const int c = 2 * ((rl + 32 * p) ^ s);
      float2 lo = {ra[p].x, ra[p].y};
      float2 hi = {ra[p].z, ra[p].w};
      *(float2*)(&As[buf][(kp0 + 0) * P2 + c]) = lo;
      *(float2*)(&As[buf][(kp0 + 1) * P2 + c]) = hi;
    }
  };

  auto loadB = [&](int k0) {
    if (TRANSB == 0) {
#pragma unroll
      for (int p = 0; p < 2; ++p) {
        const int kp = kr + 8 * p;
        rb0[p] = *(const float4*)(Bg + (long)(k0 + 2 * kp + 0) * ldb + tileN + n4);
        rb1[p] = *(const float4*)(Bg + (long)(k0 + 2 * kp + 1) * ldb + tileN + n4);
      }
    } else {
#pragma unroll
      for (int p = 0; p < 4; ++p)
        ra[p] = ra[p];  // placeholder (unused path below uses rbT)
    }
  };
  // TRANSB=1 stages into rbT (same pattern as A loader)
  float4 rbT[4];
  auto loadBT = [&](int k0) {
#pragma unroll
    for (int p = 0; p < 4; ++p)
      rbT[p] = *(const float4*)(Bg + (long)(tileN + rl + 32 * p) * ldb + k0 + k4);
  };

  auto storeB = [&](int buf) {
    if (TRANSB == 0) {
#pragma unroll
      for (int p = 0; p < 2; ++p) {
        const int kp  = kr + 8 * p;
        const int base = kp * P2 + 2 * (n4 ^ SW(kp));
        float4 w0 = {rb0[p].x, rb1[p].x, rb0[p].y, rb1[p].y};
        float4 w1 = {rb0[p].z, rb1[p].z, rb0[p].w, rb1[p].w};
        *(float4*)(&Bs[buf][base + 0]) = w0;  // 16B aligned, conflict-free
        *(float4*)(&Bs[buf][base + 4]) = w1;
      }
    } else {
      const int s = SW(kp0);
#pragma unroll
      for (int p = 0; p < 4; ++p) {
        const int c = 2 * ((rl + 32 * p) ^ s);
        float2 lo = {rbT[p].x, rbT[p].y};
        float2 hi = {rbT[p].z, rbT[p].w};
        *(float2*)(&Bs[buf][(kp0 + 0) * P2 + c]) = lo;
        *(float2*)(&Bs[buf][(kp0 + 1) * P2 + c]) = hi;
      }
    }
  };

  auto compute = [&](int buf) {
#pragma unroll
    for (int kk = 0; kk < 32; kk += 4) {
      const int s  = SW(kk >> 1);           // uniform across wave for this step
      const int pr = (kk >> 1) + lhi;       // lanes 16-31 read pair-row +1 (K+2,K+3)
      v2f af[4], bf[2];
#pragma unroll
      for (int i = 0; i < 4; ++i) {
        const int col = (wm * 64 + i * 16 + l15) ^ s;
        af[i] = *(const v2f*)(&As[buf][pr * P2 + 2 * col]);
      }
#pragma unroll
      for (int j = 0; j < 2; ++j) {
        const int col = (wn * 32 + j * 16 + l15) ^ s;
        bf[j] = *(const v2f*)(&Bs[buf][pr * P2 + 2 * col]);
      }
#pragma unroll
      for (int i = 0; i < 4; ++i)
#pragma unroll
        for (int j = 0; j < 2; ++j)
          acc[i][j] = __builtin_amdgcn_wmma_f32_16x16x4_f32(
              false, af[i], false, bf[j], (short)0, acc[i][j], false, false);
    }
  };

  // ---- software pipeline: load(t+1) | compute(t) | store(t+1) | barrier ----
  loadA(0);
  if (TRANSB == 0) loadB(0); else loadBT(0);
  storeA(0); storeB(0);
  __syncthreads();

  const int ktiles = K >> 5;
  for (int kt = 0; kt < ktiles; ++kt) {
    const int cur = kt & 1;
    if (kt + 1 < ktiles) {
      const int k0n = (kt + 1) << 5;
      loadA(k0n);
      if (TRANSB == 0) loadB(k0n); else loadBT(k0n);
      if (kt + 2 < ktiles) {                // warm L2/WGP$ for t+2 (global_prefetch_b8)
        __builtin_prefetch(Ag + (long)(tileM + rl) * lda + ((kt + 2) << 5) + k4, 0, 0);
        __builtin_prefetch(Bg + (TRANSB == 0
                                     ? (long)(((kt + 2) << 5) + 2 * kr) * ldb + tileN + n4
                                     : (long)(tileN + rl) * ldb + ((kt + 2) << 5) + k4),
                           0, 0);
      }
    }
    compute(cur);
    if (kt + 1 < ktiles) { storeA(cur ^ 1); storeB(cur ^ 1); }
    __syncthreads();
  }

  // ---- store C (VGPR v -> row v + 8*(lane>=16); lane&15 -> col) ----
  const int rhalf = lhi * 8;
#pragma unroll
  for (int i = 0; i < 4; ++i) {
#pragma unroll
    for (int j = 0; j < 2; ++j) {
      const int row0 = tileM + wm * 64 + i * 16 + rhalf;
      const int col  = tileN + wn * 32 + j * 16 + l15;
#pragma unroll
      for (int v = 0; v < 8; ++v)
        Cg[(long)(row0 + v) * ldc + col] = acc[i][j][v];
    }
  }
}

// ---------------------------------------------------------------------------
// Launch: out = [xz (1024 floats)] ++ [z2 (B*C*N*N floats)]
// Workspace: w (B*N*M) then z1 (B*C*N*M), all fp32.
// ---------------------------------------------------------------------------
extern "C" void kernel_launch(void* const* d_in, const int* in_sizes, int n_in,
                              void* d_out, int out_size, void* d_ws, size_t ws_size,
                              hipStream_t stream) {
  const float* xz = (const float*)d_in[0];   // (M, 1)
  const float* z  = (const float*)d_in[1];   // (B, C, M, M)
  const float* x  = (const float*)d_in[2];   // (B, N, 1)
  const float* ls = (const float*)d_in[3];   // scalar
  float* out = (float*)d_out;

  float* w  = (float*)d_ws;                          // B*N*M floats (16 MB)
  float* z1 = w + (size_t)BATCH * NN * MM;           // B*C*N*M floats (128 MB)

  // output[0:1024] = xz (passthrough)
  hipMemcpyAsync(out, xz, (size_t)in_sizes[0] * sizeof(float),
                 hipMemcpyDeviceToDevice, stream);

  // 1) RBF weights
  const int total = BATCH * NN * MM;
  rbf_weights_kernel<<<total / 256, 256, 0, stream>>>(x, xz, ls, w);

  // 2) z1[b,c] = w[b] (NxM) @ z[b,c] (MxM)
  dim3 grid1(MM / 128, NN / 128, BATCH * CCH);
  wmma_gemm_f32<0><<<grid1, 256, 0, stream>>>(
      w, z, z1, MM, MM, MM, MM,
      (long)NN * MM, CCH, (long)MM * MM, (long)NN * MM);

  // 3) z2[b,c] = z1[b,c] @ z1[b,c]^T  -> out + 1024
  dim3 grid2(NN / 128, NN / 128, BATCH * CCH);
  wmma_gemm_f32<1><<<grid2, 256, 0, stream>>>(
      z1, z1, out + in_sizes[0], MM, MM, NN, MM,
      (long)NN * MM, 1, (long)NN * MM, (long)NN * NN);
}